// SineGen_62998580298185
// MI455X (gfx1250) — compile-verified
//
#include <hip/hip_runtime.h>
#include <stdint.h>

// ---------------- problem constants (from reference) ----------------
#define SGN_B      8
#define SGN_T      512
#define SGN_UPP    512                         // k bits: 9
#define SGN_DIM    9
#define SGN_L      (SGN_T * SGN_UPP)           // 262144 = 2^18
#define SGN_NTAB   (SGN_B * SGN_T * SGN_DIM)   // 36864 entries
#define SGN_NTOT   (SGN_B * SGN_L * SGN_DIM)   // 18874368
#define SGN_SR     48000.0f
#define SGN_AMP    0.1f
#define SGN_NSTD   0.003f

#define TPB        256
#define STAGE_ELTS (TPB * 4)                   // 1024 floats (4 KB) per stage
#define NSTAGES    8
#define CHUNK      (STAGE_ELTS * NSTAGES)      // 8192 -> 2304 blocks exactly

// ---------------- CDNA5 async global->LDS path (gfx1250) ----------------
// vdst = VGPR holding LDS byte address, vaddr = 32-bit byte offset (GVS mode),
// saddr = 64-bit SGPR base.  Tracked by ASYNCcnt.
__device__ __forceinline__ void sgn_async_load_b128(uint64_t gbase,
                                                    uint32_t voff,
                                                    uint32_t ldsaddr) {
  asm volatile("global_load_async_to_lds_b128 %0, %1, %2"
               :: "v"(ldsaddr), "v"(voff), "s"(gbase)
               : "memory");
}
__device__ __forceinline__ void sgn_wait_async0() {
  asm volatile("s_wait_asynccnt 0x0" ::: "memory");
}
__device__ __forceinline__ void sgn_wait_async1() {
  asm volatile("s_wait_asynccnt 0x1" ::: "memory");
}
__device__ __forceinline__ void sgn_wait_ds0() {
  asm volatile("s_wait_dscnt 0x0" ::: "memory");
}

// hardware v_sin_f32 takes the angle in revolutions: sin(2*pi*x)
__device__ __forceinline__ float sgn_sin2pi(float x) {
#if __has_builtin(__builtin_amdgcn_sinf)
  return __builtin_amdgcn_sinf(x);
#else
  return __sinf(6.28318530717958647692f * x);
#endif
}
__device__ __forceinline__ float sgn_fracf(float x) { return x - floorf(x); }

// ---------------- kernel 1: fused frame-rate table (tiny) ----------------
// tab[(b*T+t)*9+d] = { r      : wrapped per-sample phase increment,
//                      P      : (sum_{t'<t} upp*r[t']) mod 1  (fp64 scan),
//                      suv    : SINE_AMP * uv(b,t),
//                      namp   : uv ? NOISE_STD : SINE_AMP/3 }
__global__ void sgn_frames(const float* __restrict__ f0,
                           const float* __restrict__ rand_ini,
                           float4* __restrict__ tab) {
  const int idx = threadIdx.x;
  if (idx >= SGN_B * SGN_DIM) return;
  const int b = idx / SGN_DIM;
  const int d = idx - b * SGN_DIM;
  const float mult = (float)(d + 1);
  double pref = 0.0;
  for (int t = 0; t < SGN_T; ++t) {
    const float f = f0[b * SGN_T + t];
    float r = sgn_fracf(f * mult / SGN_SR);
    if (t == 0) r = sgn_fracf(r + rand_ini[b * SGN_DIM + d]);
    const float uv   = (f > 0.0f) ? 1.0f : 0.0f;
    const float suv  = SGN_AMP * uv;
    const float namp = (f > 0.0f) ? SGN_NSTD : (SGN_AMP / 3.0f);
    tab[(b * SGN_T + t) * SGN_DIM + d] =
        make_float4(r, (float)(pref - floor(pref)), suv, namp);
    pref += (double)SGN_UPP * (double)r;
  }
}

// ---------------- kernel 2: uv output [B, L, 1], float4 stores ----------------
__global__ void __launch_bounds__(TPB)
sgn_uv(const float* __restrict__ f0, float* __restrict__ out_uv) {
  const uint32_t n4 = (blockIdx.x * TPB + threadIdx.x) * 4u;  // grid = B*L/4
  float u[4];
#pragma unroll
  for (int j = 0; j < 4; ++j) {
    const uint32_t n = n4 + (uint32_t)j;
    const uint32_t b = n >> 18;                 // L = 2^18
    const uint32_t t = (n & (SGN_L - 1)) >> 9;
    u[j] = (f0[b * SGN_T + t] > 0.0f) ? 1.0f : 0.0f;
  }
  *(float4*)(&out_uv[n4]) = make_float4(u[0], u[1], u[2], u[3]);
}

// ---------------- kernel 3: main streaming pass ----------------
__global__ void __launch_bounds__(TPB)
sgn_main(const float* __restrict__ noise,
         const float4* __restrict__ tab,
         float* __restrict__ out_sine,
         float* __restrict__ out_noise) {
  __shared__ __align__(16) float lbuf[2][STAGE_ELTS];  // 8 KB double buffer
  const uint32_t tid    = threadIdx.x;
  const uint32_t chunk0 = (uint32_t)blockIdx.x * (uint32_t)CHUNK;
  const uint64_t gbase  = (uint64_t)(uintptr_t)(noise + chunk0);
  const uint32_t voff0  = tid * 16u;
  const uint32_t lds_lo = (uint32_t)(uintptr_t)(&lbuf[0][tid * 4]);
  const uint32_t lds_hi = (uint32_t)(uintptr_t)(&lbuf[1][tid * 4]);

  // prologue: stage 0 in flight
  sgn_async_load_b128(gbase, voff0, lds_lo);

#pragma unroll
  for (int s = 0; s < NSTAGES; ++s) {
    if (s + 1 < NSTAGES) {
      if (s >= 1) sgn_wait_ds0();  // ds_load of the buffer being reused is done
      sgn_async_load_b128(gbase, voff0 + (uint32_t)((s + 1) * STAGE_ELTS * 4),
                          ((s + 1) & 1) ? lds_hi : lds_lo);
      sgn_wait_async1();           // in-order completion => stage s has landed
    } else {
      sgn_wait_async0();
    }

    const float4 nz4 = *(const float4*)((s & 1) ? &lbuf[1][tid * 4]
                                                : &lbuf[0][tid * 4]);
    const float nz[4] = {nz4.x, nz4.y, nz4.z, nz4.w};
    const uint32_t nbase = chunk0 + (uint32_t)(s * STAGE_ELTS) + tid * 4u;

    float sv[4], nv[4];
#pragma unroll
    for (int j = 0; j < 4; ++j) {
      const uint32_t n  = nbase + (uint32_t)j;
      const uint32_t q  = n / 9u;              // = b*L + l   (one magic-div)
      const uint32_t d  = n - q * 9u;
      const uint32_t fo = q >> 9;              // = b*T + t   (b*L>>9 = b*T)
      const uint32_t k  = q & (SGN_UPP - 1);   // = l & 511
      const float4 e = tab[fo * SGN_DIM + d];  // {r, P, 0.1*uv, namp}; L2-hot
      float ph = e.y + (float)(k + 1) * e.x;   // phase mod 1 (shifts are ints)
      ph -= floorf(ph);
      const float noi = e.w * nz[j];
      nv[j] = noi;
      sv[j] = sgn_sin2pi(ph) * e.z + noi;
    }
    *(float4*)(&out_sine[nbase])  = make_float4(sv[0], sv[1], sv[2], sv[3]);
    *(float4*)(&out_noise[nbase]) = make_float4(nv[0], nv[1], nv[2], nv[3]);
  }
}

// ---------------- launcher ----------------
extern "C" void kernel_launch(void* const* d_in, const int* in_sizes, int n_in,
                              void* d_out, int out_size, void* d_ws, size_t ws_size,
                              hipStream_t stream) {
  const float* f0       = (const float*)d_in[0];   // [8,512]
  const float* rand_ini = (const float*)d_in[1];   // [8,9]
  const float* noise    = (const float*)d_in[2];   // [8,262144,9]
  // d_in[3] = upp (compile-time constant 512)

  float4* tab = (float4*)d_ws;                     // 576 KB of ws

  float* out_sine  = (float*)d_out;                // [8,262144,9]
  float* out_uv    = out_sine + SGN_NTOT;          // [8,262144,1]
  float* out_noise = out_uv + (SGN_B * SGN_L);     // [8,262144,9]

  sgn_frames<<<1, 128, 0, stream>>>(f0, rand_ini, tab);
  sgn_uv<<<(SGN_B * SGN_L) / (TPB * 4), TPB, 0, stream>>>(f0, out_uv);
  sgn_main<<<SGN_NTOT / CHUNK, TPB, 0, stream>>>(noise, tab, out_sine, out_noise);
}